// HungarianReorder_41016937676989
// MI455X (gfx1250) — compile-verified
//
#include <hip/hip_runtime.h>
#include <hip/hip_bf16.h>
#include <stdint.h>

#define BATCH 512
#define NDIM  256

typedef __attribute__((ext_vector_type(2))) float v2f;
typedef __attribute__((ext_vector_type(8))) float v8f;

// ---------------------------------------------------------------------------
// Kernel A: column sums of D1 and D2 via V_WMMA_F32_16X16X4_F32.
// A-matrix = all ones  =>  D[m][n] = sum_k B[k][n]  (column sums).
// Each wave handles 16 columns of one (batch, matrix); 64 WMMAs of K=4 chunks.
// 16384 wave-tasks -> 2048 blocks x 8 waves.
// ---------------------------------------------------------------------------
__global__ __launch_bounds__(256) void colsum_wmma_kernel(
    const float* __restrict__ D1, const float* __restrict__ D2,
    float* __restrict__ sumA, float* __restrict__ sumB) {
  const unsigned lane = threadIdx.x & 31u;
  const unsigned wave = threadIdx.x >> 5;
  const unsigned task = blockIdx.x * 8u + wave;   // 0 .. 16383
  const unsigned b    = task >> 5;                // batch
  const unsigned rem  = task & 31u;
  const unsigned mat  = rem >> 4;                 // 0 -> D1, 1 -> D2
  const unsigned colbase = (rem & 15u) << 4;      // 16 columns per wave

  const float* __restrict__ src = (mat == 0u) ? D1 : D2;
  float* __restrict__ dst       = (mat == 0u) ? sumA : sumB;

  const unsigned col  = colbase + (lane & 15u);
  const unsigned koff = (lane >> 4) << 1;         // lanes 0-15: rows k,k+1 ; 16-31: k+2,k+3
  const float* p = src + (size_t)b * NDIM * NDIM + (size_t)koff * NDIM + col;

  v2f av; av.x = 1.0f; av.y = 1.0f;               // ones A-matrix (16x4)
  v8f c = {};
  #pragma unroll 4
  for (int k0 = 0; k0 < NDIM; k0 += 4) {
    v2f bv;
    bv.x = p[0];                                  // B 4x16 fp32 tile (2 VGPRs)
    bv.y = p[NDIM];
    c = __builtin_amdgcn_wmma_f32_16x16x4_f32(
        /*neg_a=*/false, av, /*neg_b=*/false, bv,
        /*c_mod=*/(short)0, c, /*reuse_a=*/false, /*reuse_b=*/false);
    p += 4 * NDIM;
  }
  // D row M=0 lives in accumulator VGPR0 for lanes 0..15 (N = lane).
  if (lane < 16u) dst[(size_t)b * NDIM + col] = c[0];
}

// ---------------------------------------------------------------------------
// Kernel B: per-batch stable lexsort (type, value) on both sides -> perm.
// 64-bit order-preserving key + original-index tiebreak == stable sort.
// ---------------------------------------------------------------------------
__device__ __forceinline__ unsigned long long make_key(int type, float v) {
  unsigned tv = (unsigned)type ^ 0x80000000u;                 // int -> ordered u32
  unsigned fb = __float_as_uint(v);                           // float -> ordered u32
  fb = (fb & 0x80000000u) ? ~fb : (fb | 0x80000000u);
  return ((unsigned long long)tv << 32) | (unsigned long long)fb;
}

__device__ __forceinline__ void bitonic256(unsigned long long* key, unsigned* idx) {
  const unsigned i = threadIdx.x;
  for (unsigned k = 2; k <= 256; k <<= 1) {
    for (unsigned j = k >> 1; j > 0; j >>= 1) {
      __syncthreads();
      const unsigned ixj = i ^ j;
      if (ixj > i) {
        unsigned long long ki = key[i], kj = key[ixj];
        unsigned ii = idx[i], ij = idx[ixj];
        const bool greater = (ki > kj) || (ki == kj && ii > ij);
        const bool up = ((i & k) == 0u);
        if (greater == up) {
          key[i] = kj; key[ixj] = ki;
          idx[i] = ij; idx[ixj] = ii;
        }
      }
    }
  }
  __syncthreads();
}

__global__ __launch_bounds__(256) void sort_perm_kernel(
    const float* __restrict__ sumA, const float* __restrict__ sumB,
    const int* __restrict__ types1, const int* __restrict__ types2,
    int* __restrict__ perm, float* __restrict__ typesOut) {
  const int b = blockIdx.x;
  const int t = threadIdx.x;
  __shared__ unsigned long long key[256];
  __shared__ unsigned idx[256];
  __shared__ unsigned short ord1[256];
  __shared__ unsigned short permL[256];

  // side 1: order1 = lexsort((a_row, types1))
  key[t] = make_key(types1[b * NDIM + t], sumA[b * NDIM + t]);
  idx[t] = (unsigned)t;
  bitonic256(key, idx);
  ord1[t] = (unsigned short)idx[t];

  // side 2: order2 = lexsort((b_row, types2))
  key[t] = make_key(types2[b * NDIM + t], sumB[b * NDIM + t]);
  idx[t] = (unsigned)t;
  bitonic256(key, idx);          // internal leading barrier covers key/idx rewrite

  // perm[order2[k]] = order1[k]
  permL[idx[t]] = ord1[t];
  __syncthreads();

  perm[b * NDIM + t] = (int)permL[t];
  typesOut[b * NDIM + t] = (float)types1[b * NDIM + (int)permL[t]];
}

// ---------------------------------------------------------------------------
// Kernel C: out[b,r,c] = D1[b, perm[r], perm[c]].
// Stage each source row into LDS (coalesced global side; CDNA5 async-to-LDS
// path when the builtin exists), scrambled gather happens in LDS.
// ---------------------------------------------------------------------------
#if defined(__has_builtin)
#if __has_builtin(__builtin_amdgcn_global_load_async_to_lds_b32) && \
    __has_builtin(__builtin_amdgcn_s_wait_asynccnt)
#define HAVE_ASYNC_LDS 1
#endif
#endif

typedef __attribute__((address_space(1))) int as1_int;
typedef __attribute__((address_space(3))) int as3_int;

__global__ __launch_bounds__(256) void gather_kernel(
    const float* __restrict__ D1, const int* __restrict__ perm,
    float* __restrict__ out) {
  const int b  = blockIdx.x;
  const int r0 = blockIdx.y * 16;
  const int t  = threadIdx.x;
  __shared__ int   permL[256];
  __shared__ float row[256];

  permL[t] = perm[b * NDIM + t];
  __syncthreads();

  const float* __restrict__ Db = D1 + (size_t)b * NDIM * NDIM;
  float* __restrict__ Ob       = out + (size_t)b * NDIM * NDIM;

  for (int r = r0; r < r0 + 16; ++r) {
    const int srcRow = permL[r];
#if defined(HAVE_ASYNC_LDS)
    __builtin_amdgcn_global_load_async_to_lds_b32(
        (as1_int*)(Db + (size_t)srcRow * NDIM + t),
        (as3_int*)&row[t],
        /*offset=*/0, /*cpol=*/0);
    __builtin_amdgcn_s_wait_asynccnt(0);
#else
    row[t] = Db[(size_t)srcRow * NDIM + t];
#endif
    __syncthreads();
    Ob[(size_t)r * NDIM + t] = row[permL[t]];
    __syncthreads();
  }
}

// ---------------------------------------------------------------------------
extern "C" void kernel_launch(void* const* d_in, const int* in_sizes, int n_in,
                              void* d_out, int out_size, void* d_ws, size_t ws_size,
                              hipStream_t stream) {
  const float* D1   = (const float*)d_in[0];
  const float* D2   = (const float*)d_in[1];
  const int* types1 = (const int*)d_in[2];
  const int* types2 = (const int*)d_in[3];

  float* outD = (float*)d_out;                               // [B, N, N]
  float* outT = outD + (size_t)BATCH * NDIM * NDIM;          // [B, N] (as float)

  float* sumA = (float*)d_ws;                                // B*N floats
  float* sumB = sumA + (size_t)BATCH * NDIM;                 // B*N floats
  int*   perm = (int*)(sumB + (size_t)BATCH * NDIM);         // B*N ints

  colsum_wmma_kernel<<<2048, 256, 0, stream>>>(D1, D2, sumA, sumB);
  sort_perm_kernel<<<BATCH, 256, 0, stream>>>(sumA, sumB, types1, types2, perm, outT);
  dim3 grid(BATCH, NDIM / 16);
  gather_kernel<<<grid, 256, 0, stream>>>(D1, perm, outD);
}